// Conv_block_49331994362308
// MI455X (gfx1250) — compile-verified
//
#include <hip/hip_runtime.h>

typedef __attribute__((ext_vector_type(2))) float v2f;
typedef __attribute__((ext_vector_type(8))) float v8f;

#define XN_IN   128
#define XE_IN   64
#define XN_OUT  128
#define KNEI    16
#define KTOT    320      // 128 (x) + 128 (xnj) + 64 (xej)
#define KSTEPS  80       // KTOT / 4  (wmma f32 K=4)
#define NTILES  8        // 128 output cols / 16
#define APAD    324      // 320 + 4 pad -> conflict-free ds_load_b64 for A frags

// ---------------------------------------------------------------------------
// Pack W^T = [Wc | Wn | We]^T (320 x 128) into WMMA B-fragment order:
//   frag index v = ((ntile*KSTEPS + kk)*32 + lane)*2 + j
//   holds B[k][n] with k = 4*kk + 2*(lane>>4) + j,  n = 16*ntile + (lane&15)
// (16x16x4 f32 B layout: lanes 0-15 -> K,K+1 of column n; lanes 16-31 -> K+2,K+3)
// ---------------------------------------------------------------------------
__global__ __launch_bounds__(256) void pack_weights(
    const float* __restrict__ Wc, const float* __restrict__ Wn,
    const float* __restrict__ We, float* __restrict__ Bp)
{
    int v = blockIdx.x * 256 + threadIdx.x;          // 0 .. 40959
    if (v >= NTILES * KSTEPS * 32 * 2) return;
    int j   = v & 1;
    int l   = (v >> 1) & 31;
    int rem = v >> 6;
    int kk  = rem % KSTEPS;
    int t   = rem / KSTEPS;
    int k = (kk << 2) + ((l >> 4) << 1) + j;
    int n = (t << 4) + (l & 15);
    float val;
    if (k < 128)      val = Wc[n * 128 + k];
    else if (k < 256) val = Wn[n * 128 + (k - 128)];
    else              val = We[n * 64  + (k - 256)];
    Bp[v] = val;
}

// ---------------------------------------------------------------------------
// Fused: build A = [x | mean(x[nh]) | mean(e)] (16 x 320) tile in LDS, then
// 8 waves each run 80x v_wmma_f32_16x16x4_f32 against cache-resident B frags.
// ---------------------------------------------------------------------------
__global__ __launch_bounds__(256) void fused_gnn_tile(
    const float* __restrict__ x,  const float* __restrict__ e,
    const int*   __restrict__ ij, const float* __restrict__ Bp,
    float* __restrict__ out, int Ntotal)
{
    __shared__ float As[16][APAD];
    __shared__ int   nh[16][KNEI];

    const int  tid  = threadIdx.x;
    const long row0 = (long)blockIdx.x * 16;

    // ---- Phase 1: build fused A tile -------------------------------------
    {
        // neighbor indices (one per thread)
        int r = tid >> 4, k = tid & 15;
        long rr = row0 + r; if (rr >= Ntotal) rr = Ntotal - 1;
        nh[r][k] = ij[(rr * KNEI + k) * 2];          // ij[:, :, 0]
    }
    // x part: cols [0,128)
    for (int o = tid; o < 16 * XN_IN; o += 256) {
        int r = o >> 7, c = o & 127;
        long rr = row0 + r; if (rr >= Ntotal) rr = Ntotal - 1;
        As[r][c] = x[rr * XN_IN + c];
    }
    __syncthreads();                                  // nh visible

    // gathered-neighbor mean: cols [128,256)  (x is L2-resident -> cheap)
    for (int o = tid; o < 16 * XN_IN; o += 256) {
        int r = o >> 7, c = o & 127;
        float s = 0.f;
        #pragma unroll
        for (int k = 0; k < KNEI; ++k)
            s += x[(long)nh[r][k] * XN_IN + c];
        As[r][XN_IN + c] = s * 0.0625f;
    }
    // edge mean: cols [256,320)  -- non-temporal: e streams once (410 MB),
    // keep it out of L2 so x + weights stay resident.
    for (int o = tid; o < 16 * XE_IN; o += 256) {
        int r = o >> 6, c = o & 63;
        long rr = row0 + r; if (rr >= Ntotal) rr = Ntotal - 1;
        const float* ep = e + (rr * KNEI) * XE_IN + c;
        float s = 0.f;
        #pragma unroll
        for (int k = 0; k < KNEI; ++k)
            s += __builtin_nontemporal_load(ep + k * XE_IN);
        As[r][2 * XN_IN + c] = s * 0.0625f;
    }
    __syncthreads();

    // ---- Phase 2: WMMA GEMM, one 16x16 output tile per wave --------------
    const int wave  = tid >> 5;
    const int lane  = tid & 31;
    const int mrow  = lane & 15;            // A fragment row
    const int khalf = (lane >> 4) << 1;     // 0 for lanes 0-15, 2 for 16-31
    const float* bfrag = Bp + (size_t)(wave * KSTEPS) * 64;  // 32 lanes * 2 f32

    v8f acc = {};
    #pragma unroll 4
    for (int kk = 0; kk < KSTEPS; ++kk) {
        v2f a = *(const v2f*)&As[mrow][(kk << 2) + khalf];          // ds_load_2addr_b64
        v2f b = *(const v2f*)(bfrag + (size_t)kk * 64 + lane * 2);  // global b64, L2-hot
        acc = __builtin_amdgcn_wmma_f32_16x16x4_f32(
                  false, a, false, b, (short)0, acc, false, false);
    }

    // relu + store (C/D layout: VGPR v, lanes 0-15 -> M=v, lanes 16-31 -> M=v+8)
    const int ncol = (wave << 4) + (lane & 15);
    const int mofs = (lane >> 4) << 3;
    float* obase = out + (row0 + mofs) * XN_OUT + ncol;

    if (row0 + 16 <= Ntotal) {
        // full tile: single base address, immediate-offset stores (clauseable)
        #pragma unroll
        for (int vv = 0; vv < 8; ++vv)
            obase[(size_t)vv * XN_OUT] = fmaxf(acc[vv], 0.f);
    } else {
        // only the last (partial) tile takes the guarded path
        #pragma unroll
        for (int vv = 0; vv < 8; ++vv) {
            if (row0 + mofs + vv < Ntotal)
                obase[(size_t)vv * XN_OUT] = fmaxf(acc[vv], 0.f);
        }
    }
}

// ---------------------------------------------------------------------------
extern "C" void kernel_launch(void* const* d_in, const int* in_sizes, int n_in,
                              void* d_out, int out_size, void* d_ws, size_t ws_size,
                              hipStream_t stream)
{
    const float* x  = (const float*)d_in[0];
    const float* e  = (const float*)d_in[1];
    const int*   ij = (const int*)  d_in[2];
    const float* Wc = (const float*)d_in[3];
    const float* Wn = (const float*)d_in[4];
    const float* We = (const float*)d_in[5];
    float* out = (float*)d_out;
    float* Bp  = (float*)d_ws;                 // needs 160 KB scratch

    const int N = in_sizes[0] / XN_IN;
    const int ntiles = (N + 15) / 16;

    pack_weights<<<(NTILES * KSTEPS * 64 + 255) / 256, 256, 0, stream>>>(Wc, Wn, We, Bp);
    fused_gnn_tile<<<ntiles, 256, 0, stream>>>(x, e, ij, Bp, out, N);
}